// KMeans_77670188581338
// MI455X (gfx1250) — compile-verified
//
#include <hip/hip_runtime.h>

#define NB 8
#define NPTS 65536
#define ND 64
#define NK 64
#define NITERS 25
#define BLOCKS_PER_BATCH 32
#define TILES_PER_BATCH (NPTS / 16)                        // 4096 tiles of 16 points
#define TILES_PER_BLOCK (TILES_PER_BATCH / BLOCKS_PER_BATCH) // 128
#define WAVES_PER_BLOCK 8

typedef float v2f __attribute__((ext_vector_type(2)));
typedef float v8f __attribute__((ext_vector_type(8)));

// centers0[b][k][:] = x[b][init_idx[k]][:]
__global__ void kmeans_init_centers(const float* __restrict__ x,
                                    const int* __restrict__ init_idx,
                                    float* __restrict__ centers) {
    const int bk = blockIdx.x;            // 0 .. NB*NK-1
    const int b  = bk >> 6;
    const int k  = bk & 63;
    const int d  = threadIdx.x;           // 0..63
    centers[(size_t)bk * ND + d] = x[((size_t)b * NPTS + init_idx[k]) * ND + d];
}

__global__ void kmeans_zero_ws(float* __restrict__ ws, int n) {
    const int i = blockIdx.x * blockDim.x + threadIdx.x;
    if (i < n) ws[i] = 0.0f;
}

// One iteration: assignment via f32 WMMA + LDS accumulation of per-cluster sums.
__global__ void __launch_bounds__(256)
kmeans_assign_accum(const float* __restrict__ x,
                    const float* __restrict__ centers,
                    float* __restrict__ gsums,
                    float* __restrict__ gsizes) {
    __shared__ float c_lds[NK * ND];   // centers, row-major [k][d]
    __shared__ float c2_lds[NK];       // |c_k|^2
    __shared__ float s_lds[NK * ND];   // per-block cluster sums
    __shared__ float n_lds[NK];        // per-block cluster counts

    const int b   = blockIdx.x / BLOCKS_PER_BATCH;
    const int blk = blockIdx.x % BLOCKS_PER_BATCH;
    const int tid = threadIdx.x;

    const float* __restrict__ xb = x + (size_t)b * NPTS * ND;
    const float* __restrict__ cb = centers + (size_t)b * NK * ND;

    // Load centers to LDS (4096 floats, float4 vectorized) and zero accumulators.
    for (int i = tid; i < NK * ND / 4; i += 256) {
        ((float4*)c_lds)[i] = ((const float4*)cb)[i];
        ((float4*)s_lds)[i] = make_float4(0.f, 0.f, 0.f, 0.f);
    }
    if (tid < NK) n_lds[tid] = 0.0f;
    __syncthreads();
    if (tid < NK) {
        float s = 0.0f;
        #pragma unroll 8
        for (int d = 0; d < ND; ++d) { float v = c_lds[tid * ND + d]; s += v * v; }
        c2_lds[tid] = s;
    }
    __syncthreads();

    const int wave  = tid >> 5;
    const int lane  = tid & 31;
    const int nl    = lane & 15;         // row (A) / column (B) index within half
    const int khalf = (lane >> 4) * 2;   // K sub-offset: lanes 0-15 -> K{0,1}, 16-31 -> K{2,3}

    for (int t = 0; t < TILES_PER_BLOCK / WAVES_PER_BLOCK; ++t) {
        const int tile = blk * TILES_PER_BLOCK + t * WAVES_PER_BLOCK + wave;

        // ---- A fragments: 16 points x 64 dims, sixteen 16x4 slices ----
        const float* xrow = xb + ((size_t)tile * 16 + nl) * ND + khalf;
        v2f afrag[16];
        #pragma unroll
        for (int kk = 0; kk < 16; ++kk)
            afrag[kk] = *(const v2f*)(xrow + kk * 4);

        float bestv[8];
        int   besti[8];
        #pragma unroll
        for (int r = 0; r < 8; ++r) { bestv[r] = 3.4e38f; besti[r] = 0; }

        // ---- 4 cluster tiles of 16: dot products via V_WMMA_F32_16X16X4_F32 ----
        #pragma unroll
        for (int nt = 0; nt < 4; ++nt) {
            v8f acc = {};
            const float* crow = &c_lds[(nt * 16 + nl) * ND + khalf];
            #pragma unroll
            for (int kk = 0; kk < 16; ++kk) {
                v2f bfrag = *(const v2f*)(crow + kk * 4);
                acc = __builtin_amdgcn_wmma_f32_16x16x4_f32(
                          false, afrag[kk], false, bfrag,
                          (short)0, acc, false, false);
            }
            const float c2v = c2_lds[nt * 16 + nl];
            const int   ci  = nt * 16 + nl;
            #pragma unroll
            for (int r = 0; r < 8; ++r) {
                const float sc = c2v - 2.0f * acc[r];   // x2 omitted: constant in k
                if (sc < bestv[r]) { bestv[r] = sc; besti[r] = ci; }
            }
        }

        // ---- argmin across the 16 lanes of each half (xor masks stay in-half) ----
        #pragma unroll
        for (int r = 0; r < 8; ++r) {
            #pragma unroll
            for (int m = 1; m < 16; m <<= 1) {
                const float ov = __shfl_xor(bestv[r], m, 32);
                const int   oi = __shfl_xor(besti[r], m, 32);
                if (ov < bestv[r] || (ov == bestv[r] && oi < besti[r])) {
                    bestv[r] = ov; besti[r] = oi;
                }
            }
        }

        // ---- scatter-accumulate: wave cooperatively adds each point's vector ----
        #pragma unroll
        for (int mm = 0; mm < 16; ++mm) {
            const int idx = __shfl(besti[mm & 7], (mm >> 3) * 16, 32);
            const v2f xv = *(const v2f*)(xb + ((size_t)tile * 16 + mm) * ND + lane * 2);
            atomicAdd(&s_lds[idx * ND + lane * 2 + 0], xv.x);
            atomicAdd(&s_lds[idx * ND + lane * 2 + 1], xv.y);
            if (lane == 0) atomicAdd(&n_lds[idx], 1.0f);
        }
    }
    __syncthreads();

    // ---- flush block partials to global ----
    float* gs = gsums + (size_t)b * NK * ND;
    for (int i = tid; i < NK * ND; i += 256)
        unsafeAtomicAdd(&gs[i], s_lds[i]);
    if (tid < NK)
        unsafeAtomicAdd(&gsizes[b * NK + tid], n_lds[tid]);
}

__global__ void kmeans_update_centers(const float* __restrict__ gsums,
                                      const float* __restrict__ gsizes,
                                      float* __restrict__ centers) {
    const int bk = blockIdx.x;    // 0 .. NB*NK-1
    const int d  = threadIdx.x;   // 0..63
    const float denom = fmaxf(gsizes[bk], 1.0f);
    centers[(size_t)bk * ND + d] = gsums[(size_t)bk * ND + d] / denom;
}

extern "C" void kernel_launch(void* const* d_in, const int* in_sizes, int n_in,
                              void* d_out, int out_size, void* d_ws, size_t ws_size,
                              hipStream_t stream) {
    const float* x        = (const float*)d_in[0];   // (8, 65536, 64) f32
    const int*   init_idx = (const int*)d_in[1];     // (64,) i32
    float* centers = (float*)d_out;                  // (8, 64, 64) f32, live buffer
    float* gsums   = (float*)d_ws;                   // NB*NK*ND floats
    float* gsizes  = gsums + (size_t)NB * NK * ND;   // NB*NK floats
    const int nzero = NB * NK * ND + NB * NK;

    kmeans_init_centers<<<NB * NK, ND, 0, stream>>>(x, init_idx, centers);

    for (int it = 0; it < NITERS; ++it) {
        kmeans_zero_ws<<<(nzero + 255) / 256, 256, 0, stream>>>(gsums, nzero);
        kmeans_assign_accum<<<NB * BLOCKS_PER_BATCH, 256, 0, stream>>>(
            x, centers, gsums, gsizes);
        kmeans_update_centers<<<NB * NK, ND, 0, stream>>>(gsums, gsizes, centers);
    }
}